// GINNet_36421322670663
// MI455X (gfx1250) — compile-verified
//
#include <hip/hip_runtime.h>

#define N_NODES 50000
#define N_EDGES 800000
#define N_GRAPHS 256
#define HIDDEN 128
#define N_LAYERS 5
#define OUT_DIM 10
#define BN_EPS 1e-5f

#define MTILE 128       // rows per GEMM block (8 waves x 16)
#define PITCH 132       // LDS row pitch (floats) for A tile: spreads 16-row column reads over banks
#define POOL_CHUNKS 8   // row-chunks per graph in the pooling kernel

typedef float v2f __attribute__((ext_vector_type(2)));
typedef float v8f __attribute__((ext_vector_type(8)));

// ==================== generic zero fills ====================
__global__ void zero_f32_kernel(float* __restrict__ p, int n) {
    int i = blockIdx.x * blockDim.x + threadIdx.x;
    if (i < n) p[i] = 0.f;
}
__global__ void zero_i32_kernel(int* __restrict__ p, int n) {
    int i = blockIdx.x * blockDim.x + threadIdx.x;
    if (i < n) p[i] = 0;
}

// ==================== CSR build (once per call; edges constant across layers) ====================
__global__ void deg_count_kernel(const int* __restrict__ dst, int* __restrict__ deg) {
    int e = blockIdx.x * blockDim.x + threadIdx.x;
    if (e < N_EDGES) atomicAdd(&deg[dst[e]], 1);
}

// single-block exclusive scan of deg[50000] -> rowptr[50001]; also seeds cursor
__global__ __launch_bounds__(1024) void scan_kernel(const int* __restrict__ deg,
                                                    int* __restrict__ rowptr,
                                                    int* __restrict__ cursor) {
    __shared__ int partial[1024];
    const int CHUNK = (N_NODES + 1023) / 1024;   // 49
    int t = threadIdx.x;
    int base = t * CHUNK;
    int s = 0;
    for (int i = 0; i < CHUNK; ++i) {
        int n = base + i;
        if (n < N_NODES) s += deg[n];
    }
    partial[t] = s;
    __syncthreads();
    if (t == 0) {
        int run = 0;
        for (int i = 0; i < 1024; ++i) { int v = partial[i]; partial[i] = run; run += v; }
    }
    __syncthreads();
    int run = partial[t];
    for (int i = 0; i < CHUNK; ++i) {
        int n = base + i;
        if (n < N_NODES) {
            int d = deg[n];
            rowptr[n] = run;
            cursor[n] = run;
            run += d;
        }
    }
    if (t == 1023) rowptr[N_NODES] = run;   // tail chunk is empty -> run == total
}

__global__ void csr_fill_kernel(const int* __restrict__ src, const int* __restrict__ dst,
                                int* __restrict__ cursor, int* __restrict__ esrc) {
    int e = blockIdx.x * blockDim.x + threadIdx.x;
    if (e < N_EDGES) {
        int pos = atomicAdd(&cursor[dst[e]], 1);
        esrc[pos] = src[e];
    }
}

// graph boundaries from the sorted batch vector: start[g]..start[g+1] = rows of graph g
__global__ void bounds_kernel(const int* __restrict__ batch, int* __restrict__ start) {
    int n = blockIdx.x * blockDim.x + threadIdx.x;
    if (n > N_NODES) return;
    int bcur  = (n < N_NODES) ? batch[n] : N_GRAPHS;
    int bprev = (n > 0) ? batch[n - 1] : -1;
    for (int g = bprev + 1; g <= bcur; ++g) start[g] = n;
}

// ==================== z[v] = h[v] + sum_{e in CSR(v)} h[esrc[e]]  (pure gather, no atomics) ====================
__global__ void gather_agg_kernel(const float* __restrict__ h,
                                  const int* __restrict__ rowptr,
                                  const int* __restrict__ esrc,
                                  float* __restrict__ z) {
    int v = blockIdx.x * 8 + (threadIdx.x >> 5);   // one wave per node
    if (v >= N_NODES) return;
    int lane = threadIdx.x & 31;
    int e0 = rowptr[v], e1 = rowptr[v + 1];
    float4 acc = *(const float4*)(h + (size_t)v * HIDDEN + lane * 4);
    for (int e = e0; e < e1; ++e) {
        int s = esrc[e];
        const float4 w = *(const float4*)(h + (size_t)s * HIDDEN + lane * 4);
        acc.x += w.x; acc.y += w.y; acc.z += w.z; acc.w += w.w;
    }
    *(float4*)(z + (size_t)v * HIDDEN + lane * 4) = acc;
}

// ==================== Y[M,128] = op(A)[M,128] @ W[128,128] + bias,  + fused column sum/sumsq ====================
// op(A) = A                               (fuse == 0)
// op(A) = relu(A*scale[c] + shift[c])     (fuse == 1, ss = {scale[128], shift[128]})
// statsG[0..127] += colsum(Y), statsG[128..255] += colsumsq(Y)  (statsG pre-zeroed)
__global__ __launch_bounds__(256) void gemm128_wmma_kernel(
    const float* __restrict__ A, const float* __restrict__ W,
    const float* __restrict__ bias, float* __restrict__ Y,
    const float* __restrict__ ss, int fuse, float* __restrict__ statsG)
{
    // W pair-interleaved: sW[(k/2)*256 + 2n + (k&1)] = W[k][n]  -> B frag = one b64 load
    __shared__ float sW[HIDDEN * HIDDEN];
    __shared__ float sA[MTILE * PITCH];
    __shared__ float sStats[2 * HIDDEN];

    const int t  = threadIdx.x;
    const int m0 = blockIdx.x * MTILE;

    sStats[t] = 0.f;   // blockDim == 2*HIDDEN == 256

    // stage W (coalesced global read, scattered LDS write)
    for (int idx = t; idx < HIDDEN * HIDDEN; idx += 256) {
        int k = idx >> 7, n = idx & 127;
        sW[((k >> 1) << 8) + (n << 1) + (k & 1)] = W[idx];
    }
    // stage A tile (+ optional fused BN+ReLU), coalesced float4 reads
    for (int j = 0; j < 16; ++j) {
        int q    = t + (j << 8);          // 0..4095
        int row  = q >> 5;
        int c    = (q & 31) << 2;
        int grow = m0 + row;
        float4 v = make_float4(0.f, 0.f, 0.f, 0.f);
        if (grow < N_NODES) {
            v = *(const float4*)(A + (size_t)grow * HIDDEN + c);
            if (fuse) {
                v.x = fmaxf(fmaf(v.x, ss[c + 0], ss[HIDDEN + c + 0]), 0.f);
                v.y = fmaxf(fmaf(v.y, ss[c + 1], ss[HIDDEN + c + 1]), 0.f);
                v.z = fmaxf(fmaf(v.z, ss[c + 2], ss[HIDDEN + c + 2]), 0.f);
                v.w = fmaxf(fmaf(v.w, ss[c + 3], ss[HIDDEN + c + 3]), 0.f);
            }
        }
        float* p = &sA[row * PITCH + c];
        p[0] = v.x; p[1] = v.y; p[2] = v.z; p[3] = v.w;
    }
    __syncthreads();

    const int wave = t >> 5;
    const int lane = t & 31;
    const int lLow = lane & 15;   // M (A) / N (B,C,D) within 16
    const int lHi  = lane >> 4;   // selects K pair (A/B) or row+8 (C/D)

    if (m0 + wave * 16 < N_NODES) {   // wave-uniform: EXEC stays all-1s inside
        v8f acc[8];
#pragma unroll
        for (int nt = 0; nt < 8; ++nt) { v8f zv = {}; acc[nt] = zv; }

        // A frag (ISA 16x4 f32 layout): lanes 0-15 -> K=k,k+1 ; lanes 16-31 -> K=k+2,k+3
        const float* aBase = &sA[(wave * 16 + lLow) * PITCH + (lHi << 1)];
#pragma unroll 4
        for (int k = 0; k < HIDDEN; k += 4) {
            v2f a = *(const v2f*)(aBase + k);
            const float* bBase = &sW[(((k >> 1) + lHi) << 8) + (lLow << 1)];
#pragma unroll
            for (int nt = 0; nt < 8; ++nt) {
                v2f b = *(const v2f*)(bBase + (nt << 5));
                acc[nt] = __builtin_amdgcn_wmma_f32_16x16x4_f32(
                    false, a, false, b, (short)0, acc[nt], false, false);
            }
        }

        // C/D layout: VGPR j -> row j (lanes 0-15) / row j+8 (lanes 16-31), col = lane%16
        const int rbase = m0 + wave * 16 + (lHi << 3);
#pragma unroll
        for (int nt = 0; nt < 8; ++nt) {
            int ncol = (nt << 4) + lLow;
            float bv = bias[ncol];
            float s = 0.f, sq = 0.f;
#pragma unroll
            for (int j = 0; j < 8; ++j) {
                float v = acc[nt][j] + bv;
                Y[(size_t)(rbase + j) * HIDDEN + ncol] = v;
                s += v;
                sq = fmaf(v, v, sq);
            }
            atomicAdd(&sStats[ncol], s);           // ds_add_f32
            atomicAdd(&sStats[HIDDEN + ncol], sq); // ds_add_f32
        }
    }
    __syncthreads();
    atomicAdd(&statsG[t], sStats[t]);   // 256 global atomics per block
}

// ==================== BN scale/shift from stats ====================
__global__ void scaleshift_kernel(const float* __restrict__ stats,
                                  const float* __restrict__ gamma,
                                  const float* __restrict__ beta,
                                  float* __restrict__ ss) {
    int c = threadIdx.x;
    float mean = stats[c] * (1.0f / N_NODES);
    float var  = stats[HIDDEN + c] * (1.0f / N_NODES) - mean * mean;  // biased, as torch
    float sc   = gamma[c] * rsqrtf(var + BN_EPS);
    ss[c]          = sc;
    ss[HIDDEN + c] = fmaf(-mean, sc, beta[c]);
}

// ==================== h = relu(BN(z)); pooled[g] += column-sums over graph rows ====================
// block = (graph, chunk); thread = column. One atomic per (graph, chunk, column).
__global__ void bn_relu_pool_kernel(const float* __restrict__ Z,
                                    const float* __restrict__ ss,
                                    const int* __restrict__ start,
                                    float* __restrict__ H,
                                    float* __restrict__ pooled) {
    int g     = blockIdx.x >> 3;          // POOL_CHUNKS == 8
    int chunk = blockIdx.x & 7;
    int c     = threadIdx.x;              // 128
    int r0 = start[g], r1 = start[g + 1];
    int per = ((r1 - r0) + POOL_CHUNKS - 1) >> 3;
    int a = r0 + chunk * per;
    int b = a + per; if (b > r1) b = r1;
    float sc = ss[c], sh = ss[HIDDEN + c];
    float acc = 0.f;
    for (int r = a; r < b; ++r) {
        float v = fmaxf(fmaf(Z[(size_t)r * HIDDEN + c], sc, sh), 0.f);
        H[(size_t)r * HIDDEN + c] = v;
        acc += v;
    }
    atomicAdd(&pooled[(size_t)g * HIDDEN + c], acc);
}

// ==================== head MLP: relu(concat(pooled) @ fc1 + b) @ fc2 + b ====================
__global__ void head_kernel(const float* __restrict__ pooled,
                            const float* __restrict__ fc1W, const float* __restrict__ fc1b,
                            const float* __restrict__ fc2W, const float* __restrict__ fc2b,
                            float* __restrict__ out) {
    __shared__ float f[N_LAYERS * HIDDEN];
    __shared__ float hb[HIDDEN];
    int g = blockIdx.x, t = threadIdx.x;   // 128 threads
    for (int l = 0; l < N_LAYERS; ++l)
        f[l * HIDDEN + t] = pooled[((size_t)l * N_GRAPHS + g) * HIDDEN + t];
    __syncthreads();
    float acc = fc1b[t];
    for (int j = 0; j < N_LAYERS * HIDDEN; ++j)
        acc = fmaf(f[j], fc1W[(size_t)j * HIDDEN + t], acc);
    hb[t] = fmaxf(acc, 0.f);
    __syncthreads();
    if (t < OUT_DIM) {
        float o = fc2b[t];
        for (int c = 0; c < HIDDEN; ++c)
            o = fmaf(hb[c], fc2W[c * OUT_DIM + t], o);
        out[(size_t)g * OUT_DIM + t] = o;
    }
}

extern "C" void kernel_launch(void* const* d_in, const int* in_sizes, int n_in,
                              void* d_out, int out_size, void* d_ws, size_t ws_size,
                              hipStream_t stream) {
    const float* x    = (const float*)d_in[0];
    const float* W1   = (const float*)d_in[1];
    const float* b1   = (const float*)d_in[2];
    const float* g1   = (const float*)d_in[3];
    const float* be1  = (const float*)d_in[4];
    const float* W2   = (const float*)d_in[5];
    const float* b2   = (const float*)d_in[6];
    const float* g2   = (const float*)d_in[7];
    const float* be2  = (const float*)d_in[8];
    const float* fc1W = (const float*)d_in[9];
    const float* fc1b = (const float*)d_in[10];
    const float* fc2W = (const float*)d_in[11];
    const float* fc2b = (const float*)d_in[12];
    const int*   ei   = (const int*)d_in[13];
    const int*   batch= (const int*)d_in[14];
    const int* src = ei;
    const int* dst = ei + N_EDGES;

    // -------- workspace layout --------
    const size_t NH = (size_t)N_NODES * HIDDEN;   // 6,400,000 floats
    float* z      = (float*)d_ws;
    float* y      = z + NH;
    float* h      = y + NH;
    float* pooled = h + NH;                                        // 5*256*128
    float* stats  = pooled + (size_t)N_LAYERS * N_GRAPHS * HIDDEN; // 256
    float* ss     = stats + 2 * HIDDEN;                            // 256
    int*   deg    = (int*)(ss + 2 * HIDDEN);                       // 50000
    int*   rowptr = deg + N_NODES;                                 // 50001
    int*   cursor = rowptr + (N_NODES + 1);                        // 50000
    int*   start  = cursor + N_NODES;                              // 257
    int*   esrc   = start + (N_GRAPHS + 1);                        // 800000

    const int gemmBlocks = (N_NODES + MTILE - 1) / MTILE;          // 391
    const int poolN      = N_LAYERS * N_GRAPHS * HIDDEN;

    // -------- one-time per call: CSR by dst + graph boundaries + pooled zero --------
    zero_i32_kernel<<<(N_NODES + 255) / 256, 256, 0, stream>>>(deg, N_NODES);
    deg_count_kernel<<<(N_EDGES + 255) / 256, 256, 0, stream>>>(dst, deg);
    scan_kernel<<<1, 1024, 0, stream>>>(deg, rowptr, cursor);
    csr_fill_kernel<<<(N_EDGES + 255) / 256, 256, 0, stream>>>(src, dst, cursor, esrc);
    bounds_kernel<<<(N_NODES + 1 + 255) / 256, 256, 0, stream>>>(batch, start);
    zero_f32_kernel<<<(poolN + 255) / 256, 256, 0, stream>>>(pooled, poolN);

    // -------- layers --------
    for (int i = 0; i < N_LAYERS; ++i) {
        const float* hin = (i == 0) ? x : h;

        gather_agg_kernel<<<(N_NODES + 7) / 8, 256, 0, stream>>>(hin, rowptr, esrc, z);

        zero_f32_kernel<<<1, 256, 0, stream>>>(stats, 2 * HIDDEN);
        gemm128_wmma_kernel<<<gemmBlocks, 256, 0, stream>>>(
            z, W1 + (size_t)i * HIDDEN * HIDDEN, b1 + i * HIDDEN, y, nullptr, 0, stats);
        scaleshift_kernel<<<1, HIDDEN, 0, stream>>>(stats, g1 + i * HIDDEN, be1 + i * HIDDEN, ss);

        zero_f32_kernel<<<1, 256, 0, stream>>>(stats, 2 * HIDDEN);
        gemm128_wmma_kernel<<<gemmBlocks, 256, 0, stream>>>(
            y, W2 + (size_t)i * HIDDEN * HIDDEN, b2 + i * HIDDEN, z, ss, 1, stats);
        scaleshift_kernel<<<1, HIDDEN, 0, stream>>>(stats, g2 + i * HIDDEN, be2 + i * HIDDEN, ss);

        bn_relu_pool_kernel<<<N_GRAPHS * POOL_CHUNKS, HIDDEN, 0, stream>>>(
            z, ss, start, h, pooled + (size_t)i * N_GRAPHS * HIDDEN);
    }

    head_kernel<<<N_GRAPHS, HIDDEN, 0, stream>>>(pooled, fc1W, fc1b, fc2W, fc2b, (float*)d_out);
}